// GINRegressor_64347200028747
// MI455X (gfx1250) — compile-verified
//
#include <hip/hip_runtime.h>
#include <hip/hip_bf16.h>

#define N_NODES 100000
#define N_EDGES 1600000
#define NF 64
#define N_TILES (N_NODES / 16)   // 6250, exact
#define BN_EPS 1e-5f
#define WT_STRIDE 68             // padded LDS stride -> conflict-free b64 reads

typedef __attribute__((ext_vector_type(2))) float v2f;
typedef __attribute__((ext_vector_type(8))) float v8f;
typedef __attribute__((ext_vector_type(4))) unsigned int u32x4;
typedef __attribute__((ext_vector_type(4))) int i32x4;
typedef __attribute__((ext_vector_type(8))) int i32x8;

#if __has_builtin(__builtin_amdgcn_tensor_load_to_lds) && \
    __has_builtin(__builtin_amdgcn_s_wait_tensorcnt)
#define GIN_TDM 1
#endif

#ifdef GIN_TDM
// ---------------------------------------------------------------------------
// TDM: DMA a 64x64 f32 tensor (16KB) from global into LDS, row-major.
// D# per ISA 8.3/8.4: 2D tensor, data_size=4B (code 2), tile == tensor,
// dim0_stride = 64 elements. Groups 2/3 zero (<=2D). Issued by one wave;
// completion via s_wait_tensorcnt 0.
// This toolchain exposes the 6-arg builtin:
//   (u32x4 g0, i32x8 g1, i32x4, i32x4, i32x8, i32 cpol)
// ---------------------------------------------------------------------------
__device__ __forceinline__ void tdm_load_w64(const float* gsrc, unsigned lds_byte_addr) {
    unsigned long long ga = (unsigned long long)gsrc;
    u32x4 g0;
    g0[0] = 1u;                                            // count=1, user desc
    g0[1] = lds_byte_addr;                                 // lds_addr
    g0[2] = (unsigned)ga;                                  // global_addr[31:0]
    g0[3] = (unsigned)((ga >> 32) & 0x1FFFFFFu) | (2u << 30); // addr[56:32], type=2
    i32x8 g1;
    g1[0] = (int)(2u << 16);   // workgroup_mask=0, data_size=2 (4 bytes)
    g1[1] = (int)(64u << 16);  // tensor_dim0 = 64 (low 16 bits at [63:48])
    g1[2] = (int)(64u << 16);  // tensor_dim0 hi=0, tensor_dim1 = 64
    g1[3] = (int)(64u << 16);  // tensor_dim1 hi=0, tile_dim0 = 64
    g1[4] = 64;                // tile_dim1 = 64, tile_dim2 = 0
    g1[5] = 64;                // tensor_dim0_stride = 64
    g1[6] = 0;
    g1[7] = 0;
    i32x4 z4 = {0, 0, 0, 0};
    i32x8 z8 = {0, 0, 0, 0, 0, 0, 0, 0};
    __builtin_amdgcn_tensor_load_to_lds(g0, g1, z4, z4, z8, 0);
    __builtin_amdgcn_s_wait_tensorcnt(0);
}
#endif

// ---------------------------------------------------------------------------
// init: s = h (so the scatter-add produces h + agg directly); zero BN stats
// ---------------------------------------------------------------------------
__global__ void gin_init_kernel(const float* __restrict__ h,
                                float* __restrict__ s,
                                float* __restrict__ stats) {
    if (blockIdx.x == 0 && threadIdx.x < 128) stats[threadIdx.x] = 0.0f;
    size_t i = (size_t)blockIdx.x * 256 + threadIdx.x;   // float4 index
    if (i < (size_t)N_NODES * (NF / 4)) {
        ((float4*)s)[i] = ((const float4*)h)[i];
    }
}

// ---------------------------------------------------------------------------
// scatter: s[dst][f] += h[src][f] for every edge. One thread per (edge,feat):
// 64 consecutive threads share an edge -> coalesced 256B gather + scatter.
// ---------------------------------------------------------------------------
__global__ void gin_scatter_kernel(const float* __restrict__ h,
                                   const long long* __restrict__ ei,
                                   float* __restrict__ s) {
    int t = blockIdx.x * 256 + threadIdx.x;
    int e = t >> 6;
    int f = t & 63;
    if (e >= N_EDGES) return;
    long long src = ei[e];
    long long dst = ei[(long long)N_EDGES + e];
    unsafeAtomicAdd(&s[(size_t)dst * NF + f], h[(size_t)src * NF + f]);
}

// ---------------------------------------------------------------------------
// Stage W into LDS transposed: wT[n*WT_STRIDE + k] = W[k*64 + n].
// TDM path: DMA row-major into stage, then LDS->LDS transpose.
// ---------------------------------------------------------------------------
__device__ __forceinline__ void stage_weights(const float* __restrict__ W,
                                              float* __restrict__ wT,
                                              float* __restrict__ stage) {
    int tid = threadIdx.x;
#ifdef GIN_TDM
    if (tid < 32) tdm_load_w64(W, (unsigned)(uintptr_t)stage);
    __syncthreads();
    for (int i = tid; i < NF * NF; i += 256) {
        int k = i >> 6, n = i & 63;
        wT[n * WT_STRIDE + k] = stage[i];
    }
#else
    (void)stage;
    for (int i = tid; i < NF * NF; i += 256) {
        int k = i >> 6, n = i & 63;
        wT[n * WT_STRIDE + k] = W[i];
    }
#endif
}

// ---------------------------------------------------------------------------
// gemm1: t = s @ W1 + b1  (in-place on s), plus per-column BN stat partials.
// One wave per 16-row tile; V_WMMA_F32_16X16X4_F32, K stepped by 4.
// Fragment layout (ISA 7.12.2):
//   A: lane<16 -> (M=lane, K=k,k+1); lane>=16 -> (M=lane-16, K=k+2,k+3)
//   B: lane<16 -> (N=lane, K=k,k+1); lane>=16 -> (N=lane-16, K=k+2,k+3)
//   C: vgpr r: lane<16 -> (M=r, N=lane); lane>=16 -> (M=8+r, N=lane-16)
// B fragment = contiguous pair in transposed LDS -> single ds_load_b64.
// ---------------------------------------------------------------------------
__global__ void __launch_bounds__(256)
gin_gemm1_kernel(float* __restrict__ s,
                 const float* __restrict__ W1,
                 const float* __restrict__ b1,
                 float* __restrict__ stats) {
    __shared__ float wT[NF * WT_STRIDE];
#ifdef GIN_TDM
    __shared__ float stage[NF * NF];
    stage_weights(W1, wT, stage);
#else
    stage_weights(W1, wT, nullptr);
#endif
    __syncthreads();

    int tid = threadIdx.x;
    int wave = tid >> 5, lane = tid & 31;
    int tile = blockIdx.x * 8 + wave;
    if (tile >= N_TILES) return;
    int rbase = tile * 16;
    int lh = lane & 15;            // row (A) / column (B,C) within tile
    int kh = (lane >> 4) << 1;     // 0 or 2: K offset of this half-wave

    const float* arow = s + (size_t)(rbase + lh) * NF + kh;
    const float* brow = wT + lh * WT_STRIDE + kh;   // + nt*16*WT_STRIDE + k

    v8f acc[4] = {};
    for (int k = 0; k < NF; k += 4) {
        v2f a = *(const v2f*)(arow + k);   // (K=k+kh, K=k+kh+1) of row lh
#pragma unroll
        for (int nt = 0; nt < 4; ++nt) {
            v2f b = *(const v2f*)(brow + nt * (16 * WT_STRIDE) + k);
            acc[nt] = __builtin_amdgcn_wmma_f32_16x16x4_f32(
                false, a, false, b, (short)0, acc[nt], false, false);
        }
    }

    int rowoff = (lane >> 4) * 8;
#pragma unroll
    for (int nt = 0; nt < 4; ++nt) {
        int col = nt * 16 + lh;
        float bias = b1[col];
        float csum = 0.0f, csq = 0.0f;
#pragma unroll
        for (int r = 0; r < 8; ++r) {
            float v = acc[nt][r] + bias;
            s[(size_t)(rbase + rowoff + r) * NF + col] = v;
            csum += v;
            csq += v * v;
        }
        unsafeAtomicAdd(&stats[col], csum);        // column sums
        unsafeAtomicAdd(&stats[NF + col], csq);    // column sum-of-squares
    }
}

// ---------------------------------------------------------------------------
// BN finalize: fold batch stats + gamma/beta into affine scale/shift
// ---------------------------------------------------------------------------
__global__ void gin_bn_kernel(float* __restrict__ stats,
                              const float* __restrict__ gamma,
                              const float* __restrict__ beta) {
    int n = threadIdx.x;     // 64 threads
    const float invN = 1.0f / (float)N_NODES;
    float mu = stats[n] * invN;
    float var = stats[NF + n] * invN - mu * mu;
    float inv = rsqrtf(var + BN_EPS);
    float a = gamma[n] * inv;
    stats[128 + n] = a;                 // scale
    stats[192 + n] = beta[n] - mu * a;  // shift
}

// ---------------------------------------------------------------------------
// gemm2: out = relu( relu(t*scale+shift) @ W2 + b2 )   (layers 0..3)
// Same WMMA tiling; BN affine + ReLU applied to A-fragments on the fly.
// ---------------------------------------------------------------------------
__global__ void __launch_bounds__(256)
gin_gemm2_kernel(const float* __restrict__ t,
                 const float* __restrict__ stats,
                 const float* __restrict__ W2,
                 const float* __restrict__ b2,
                 float* __restrict__ out) {
    __shared__ float wT[NF * WT_STRIDE];
    __shared__ float sc[NF];
    __shared__ float sf[NF];
#ifdef GIN_TDM
    __shared__ float stage[NF * NF];
    stage_weights(W2, wT, stage);
#else
    stage_weights(W2, wT, nullptr);
#endif
    int tid = threadIdx.x;
    if (tid < NF) { sc[tid] = stats[128 + tid]; sf[tid] = stats[192 + tid]; }
    __syncthreads();

    int wave = tid >> 5, lane = tid & 31;
    int tile = blockIdx.x * 8 + wave;
    if (tile >= N_TILES) return;
    int rbase = tile * 16;
    int lh = lane & 15;
    int kh = (lane >> 4) << 1;

    const float* arow = t + (size_t)(rbase + lh) * NF + kh;
    const float* brow = wT + lh * WT_STRIDE + kh;

    v8f acc[4] = {};
    for (int k = 0; k < NF; k += 4) {
        v2f a = *(const v2f*)(arow + k);
        int k0 = k + kh;
        a.x = fmaxf(a.x * sc[k0] + sf[k0], 0.0f);
        a.y = fmaxf(a.y * sc[k0 + 1] + sf[k0 + 1], 0.0f);
#pragma unroll
        for (int nt = 0; nt < 4; ++nt) {
            v2f b = *(const v2f*)(brow + nt * (16 * WT_STRIDE) + k);
            acc[nt] = __builtin_amdgcn_wmma_f32_16x16x4_f32(
                false, a, false, b, (short)0, acc[nt], false, false);
        }
    }

    int rowoff = (lane >> 4) * 8;
#pragma unroll
    for (int nt = 0; nt < 4; ++nt) {
        int col = nt * 16 + lh;
        float bias = b2[col];
#pragma unroll
        for (int r = 0; r < 8; ++r) {
            float v = fmaxf(acc[nt][r] + bias, 0.0f);
            out[(size_t)(rbase + rowoff + r) * NF + col] = v;
        }
    }
}

// ---------------------------------------------------------------------------
// final layer: W2 is 64x1 -> wave-per-node dot product with shfl reduction
// ---------------------------------------------------------------------------
__global__ void gin_final_kernel(const float* __restrict__ t,
                                 const float* __restrict__ stats,
                                 const float* __restrict__ W2,  // [64]
                                 const float* __restrict__ b2,  // [1]
                                 float* __restrict__ out) {
    int wave = threadIdx.x >> 5, lane = threadIdx.x & 31;
    int node = blockIdx.x * 8 + wave;
    if (node >= N_NODES) return;
    int j = lane * 2;
    v2f tv = *(const v2f*)(t + (size_t)node * NF + j);
    float v = fmaxf(tv.x * stats[128 + j] + stats[192 + j], 0.0f) * W2[j] +
              fmaxf(tv.y * stats[128 + j + 1] + stats[192 + j + 1], 0.0f) * W2[j + 1];
#pragma unroll
    for (int off = 16; off > 0; off >>= 1) v += __shfl_xor(v, off, 32);
    if (lane == 0) out[node] = fmaxf(v + b2[0], 0.0f);
}

// ---------------------------------------------------------------------------
// Input order: d_in[0]=x, d_in[1]=edge_index(int64),
//              d_in[2+6*l+{0..5}] = l{l}.{W1,b1,gamma,beta,W2,b2}
// Workspace: bufA | bufB | bufS(sum->t) | stats(256 floats)  (~77 MB)
// ---------------------------------------------------------------------------
extern "C" void kernel_launch(void* const* d_in, const int* in_sizes, int n_in,
                              void* d_out, int out_size, void* d_ws, size_t ws_size,
                              hipStream_t stream) {
    const float* x = (const float*)d_in[0];
    const long long* ei = (const long long*)d_in[1];
    auto P = [&](int layer, int idx) -> const float* {
        return (const float*)d_in[2 + layer * 6 + idx];
    };

    float* ws = (float*)d_ws;
    const size_t NB = (size_t)N_NODES * NF;
    float* bufA  = ws;
    float* bufB  = ws + NB;
    float* bufS  = ws + 2 * NB;
    float* stats = ws + 3 * NB;   // [0:64) sum, [64:128) sumsq, [128:192) scale, [192:256) shift

    const int initBlocks    = (int)((NB / 4 + 255) / 256);            // 6250
    const int scatterBlocks = (int)(((size_t)N_EDGES * NF) / 256);    // 400000
    const int gemmBlocks    = (N_TILES + 7) / 8;                      // 782
    const int finalBlocks   = (N_NODES + 7) / 8;                      // 12500

    const float* hin = x;
    for (int l = 0; l < 5; ++l) {
        gin_init_kernel<<<initBlocks, 256, 0, stream>>>(hin, bufS, stats);
        gin_scatter_kernel<<<scatterBlocks, 256, 0, stream>>>(hin, ei, bufS);
        gin_gemm1_kernel<<<gemmBlocks, 256, 0, stream>>>(bufS, P(l, 0), P(l, 1), stats);
        gin_bn_kernel<<<1, 64, 0, stream>>>(stats, P(l, 2), P(l, 3));
        if (l < 4) {
            float* hout = (l & 1) ? bufB : bufA;
            gin_gemm2_kernel<<<gemmBlocks, 256, 0, stream>>>(bufS, stats, P(l, 4), P(l, 5), hout);
            hin = hout;
        } else {
            gin_final_kernel<<<finalBlocks, 256, 0, stream>>>(bufS, stats, P(l, 4), P(l, 5),
                                                              (float*)d_out);
        }
    }
}